// TwoBodyToSpherical_7043746365488
// MI455X (gfx1250) — compile-verified
//
#include <hip/hip_runtime.h>
#include <math.h>

// ---------------------------------------------------------------------------
// TwoBodyToSpherical: scatter feat[5800,5800] -> out[500,500,14,14] (zero-fill)
// Inverse-map gather: every output element written exactly once via coalesced
// b128 non-temporal streaming stores; index tables staged to LDS through the
// gfx1250 async global->LDS path (ASYNCcnt / s_wait_asynccnt).
// ---------------------------------------------------------------------------

#if defined(__HIP_DEVICE_COMPILE__)
#define HAS_ASYNC_LDS  __has_builtin(__builtin_amdgcn_global_load_async_to_lds_b32)
#define HAS_WAIT_ASYNC __has_builtin(__builtin_amdgcn_s_wait_asynccnt)
#else
#define HAS_ASYNC_LDS 0
#define HAS_WAIT_ASYNC 0
#endif

typedef __attribute__((ext_vector_type(4))) float v4f;

__global__ void tbts_init_inv(int* __restrict__ inv, int total) {
    int i = blockIdx.x * blockDim.x + threadIdx.x;
    if (i < total) inv[i] = -1;
}

__global__ void tbts_build_inv(const int* __restrict__ ids,
                               int* __restrict__ inv, int n_ao) {
    int i = blockIdx.x * blockDim.x + threadIdx.x;
    if (i < n_ao) inv[ids[i]] = i;   // ids are unique -> plain scatter
}

// Specialized gather: A atoms, R reps known at compile time (divs -> mul/shift).
// Block = 256 threads. blockIdx.y = a1, blockIdx.x = chunk of CHUNK a2-atoms.
// A block's output range is contiguous and 16B-aligned -> b128 stores.
template <int A, int R, int CHUNK>
__global__ __launch_bounds__(256) void tbts_gather(
    const float* __restrict__ feat,   // [n_ao, n_ao]
    const int*   __restrict__ inv,    // [A*R] : flat-id -> AO index or -1
    float*       __restrict__ out,    // [A, A, R, R]
    int n_ao) {
    constexpr int RR = R * R;         // 196: divisible by 4, *16B-aligned base
    __shared__ int sRow[R];
    __shared__ int sCol[CHUNK * R];

    const int t    = threadIdx.x;
    const int a1   = blockIdx.y;
    const int a2lo = blockIdx.x * CHUNK;
    const int cnt  = min(CHUNK, A - a2lo);
    const int colN = cnt * R;

    // ---- stage inv slices into LDS (async global->LDS on gfx1250) ----
#if HAS_ASYNC_LDS
    if (t < R) {
        __builtin_amdgcn_global_load_async_to_lds_b32(
            (int*)(inv + a1 * R + t), &sRow[t], 0, 0);
    } else if (t < R + colN) {
        const int c = t - R;
        __builtin_amdgcn_global_load_async_to_lds_b32(
            (int*)(inv + a2lo * R + c), &sCol[c], 0, 0);
    }
#if HAS_WAIT_ASYNC
    __builtin_amdgcn_s_wait_asynccnt(0);
#endif
#else
    if (t < R)               sRow[t]     = inv[a1 * R + t];
    else if (t < R + colN)   sCol[t - R] = inv[a2lo * R + (t - R)];
#endif
    __syncthreads();

    float* outBase = out + (long)(a1 * A + a2lo) * RR;  // multiple of 16B
    const int nVec = (cnt * RR) >> 2;                   // float4 count

    for (int q = t; q < nVec; q += 256) {
        const int e0 = q << 2;
        v4f v;
#pragma unroll
        for (int k = 0; k < 4; ++k) {
            const int e   = e0 + k;
            const int a2i = e / RR;          // const divisors -> mul/shift
            const int rem = e - a2i * RR;
            const int r1  = rem / R;
            const int r2  = rem - r1 * R;
            const int si  = sRow[r1];
            const int sj  = sCol[a2i * R + r2];
            float x = 0.0f;
            if ((si | sj) >= 0)              // both source indices valid
                x = __builtin_nontemporal_load(feat + (si * n_ao + sj));
            v[k] = x;
        }
        __builtin_nontemporal_store(v, (v4f*)(outBase + e0));  // b128 stream
    }
}

// Generic fallback (runtime A, R read from the device scalar inputs).
__global__ void tbts_gather_generic(
    const float* __restrict__ feat, const int* __restrict__ inv,
    const int* __restrict__ dA, const int* __restrict__ dR,
    float* __restrict__ out, int n_ao, int out_size) {
    int o = blockIdx.x * blockDim.x + threadIdx.x;
    if (o >= out_size) return;
    const int A = *dA, R = *dR;              // uniform -> scalar loads
    const int r2 = o % R;  int tt = o / R;
    const int r1 = tt % R; int u  = tt / R;
    const int a2 = u % A;  const int a1 = u / A;
    const int si = inv[a1 * R + r1];
    const int sj = inv[a2 * R + r2];
    out[o] = ((si | sj) >= 0) ? feat[si * n_ao + sj] : 0.0f;
}

extern "C" void kernel_launch(void* const* d_in, const int* in_sizes, int n_in,
                              void* d_out, int out_size, void* d_ws, size_t ws_size,
                              hipStream_t stream) {
    const float* feat = (const float*)d_in[0];
    const int*   ids  = (const int*)d_in[1];
    const int*   dA   = (const int*)d_in[2];   // n_atoms (device scalar)
    const int*   dR   = (const int*)d_in[3];   // num_reps_1d (device scalar)
    float*       out  = (float*)d_out;

    const int n_ao = in_sizes[1];

    // total = A*R = isqrt(out_size)  (out is [total x total] pre-transpose)
    int total = (int)(sqrt((double)out_size) + 0.5);
    while ((long)total * total > (long)out_size) --total;
    while ((long)(total + 1) * (total + 1) <= (long)out_size) ++total;

    int* inv = (int*)d_ws;                     // total ints of scratch

    tbts_init_inv<<<(total + 255) / 256, 256, 0, stream>>>(inv, total);
    tbts_build_inv<<<(n_ao + 255) / 256, 256, 0, stream>>>(ids, inv, n_ao);

    if (total == 7000) {
        constexpr int A = 500, R = 14, CHUNK = 32;
        dim3 grid((A + CHUNK - 1) / CHUNK, A);
        tbts_gather<A, R, CHUNK><<<grid, 256, 0, stream>>>(feat, inv, out, n_ao);
    } else {
        tbts_gather_generic<<<(out_size + 255) / 256, 256, 0, stream>>>(
            feat, inv, dA, dR, out, n_ao, out_size);
    }
}